// SelectTopk_2216203124743
// MI455X (gfx1250) — compile-verified
//
#include <hip/hip_runtime.h>
#include <stdint.h>

// MoE router top-k softmax for MI455X (gfx1250).
//  - Memory-bound reference (~352 MB @ 23.3 TB/s ~= 15 us floor); no matmul -> no WMMA.
//  - CDNA5 path: each wave32 issues its own Tensor Data Mover descriptor
//    (tensor_load_to_lds) for its 32-token slice of logits, waits on its own
//    TENSORcnt, and proceeds with NO block barrier. Descriptor-driven LDS row
//    padding (+4 DWORDs per 64) kills bank conflicts on the per-token reads.
//  - Renormalization cancels the full-softmax denominator:
//      w_k = exp(l_k - max) / sum_{j in top8} exp(l_j - max)
//    so only the top-8 logits hit the exp unit.
//  - Selection: per-thread 8 argmax rounds over order-preserving u32 keys,
//    split into 4 independent chains for VOPD/co-exec ILP.

#define NUM_EXPERTS 64
#define TOPK 8
#define TPB 128                       // tokens (threads) per block, 4 wave32s
#define SLICE_TOKENS 32               // tokens per wave (one TDM descriptor each)
#define SLICE_ELEMS (SLICE_TOKENS * NUM_EXPERTS)  // 2048 fp32
#define ROW_DW (NUM_EXPERTS + 4)      // 68 DWORDs: 64 data + 4 pad (TDM pad_enable)

typedef uint32_t u32x4 __attribute__((ext_vector_type(4)));
typedef uint32_t u32x8 __attribute__((ext_vector_type(8)));

// Order-preserving, exactly invertible fp32 -> u32 key.
__device__ __forceinline__ uint32_t f32_to_key(float f) {
  uint32_t u = __float_as_uint(f);
  return u ^ ((uint32_t)((int32_t)u >> 31) | 0x80000000u);
}
__device__ __forceinline__ float key_to_f32(uint32_t k) {
  uint32_t mask = (uint32_t)((int32_t)(~k) >> 31) | 0x80000000u;
  return __uint_as_float(k ^ mask);
}

__global__ __launch_bounds__(TPB) void topk_softmax_kernel(
    const float* __restrict__ logits, float* __restrict__ w_out,
    int* __restrict__ id_out, int* __restrict__ src_out, long num_tokens) {
  __shared__ __align__(16) float smem[TPB * ROW_DW];  // 34,816 B

  const long block_tok0 = (long)blockIdx.x * TPB;

  // ---- per-wave TDM: async DMA of this wave's [32 x 64] fp32 slice into LDS ----
  {
    const int wave = __builtin_amdgcn_readfirstlane((int)(threadIdx.x >> 5));
    const long slice_tok0 = block_tok0 + (long)wave * SLICE_TOKENS;
    const uint64_t gaddr =
        (uint64_t)(uintptr_t)(logits + slice_tok0 * NUM_EXPERTS);
    const uint32_t lds_base =
        (uint32_t)(uintptr_t)&smem[0] + (uint32_t)wave * SLICE_TOKENS * ROW_DW * 4u;
    long remaining = (num_tokens - slice_tok0) * NUM_EXPERTS;
    uint32_t nelem = remaining >= (long)SLICE_ELEMS ? (uint32_t)SLICE_ELEMS
                     : (remaining < 0 ? 0u : (uint32_t)remaining);
    // D# group 0 (4 SGPRs): count=1 | lds_addr | global_addr[56:0] | type=2
    u32x4 g0;
    g0[0] = 1u;
    g0[1] = lds_base;
    g0[2] = (uint32_t)gaddr;
    g0[3] = ((uint32_t)(gaddr >> 32) & 0x01FFFFFFu) | (2u << 30);
    // D# group 1 (8 SGPRs): data_size=4B, pad 4 DWORDs after every 64 DWORDs,
    // tensor_dim0 = valid elements (OOB reads -> 0), tile_dim0 = SLICE_ELEMS.
    u32x8 g1;
    g1[0] = (2u << 16)        // data_size = 4 bytes
          | (1u << 20)        // pad_enable
          | (5u << 22)        // pad_interval: 64 DWORDs
          | (3u << 25);       // pad_amount:   4 DWORDs
    g1[1] = (nelem & 0xFFFFu) << 16;             // tensor_dim0[15:0] @ bit48
    g1[2] = (nelem >> 16) | (1u << 16);          // tensor_dim0[31:16] | tensor_dim1=1
    g1[3] = ((uint32_t)SLICE_ELEMS) << 16;       // tile_dim0 @ bit112
    g1[4] = 0u;                                  // tile_dim1=0, tile_dim2=0
    g1[5] = (uint32_t)SLICE_ELEMS;               // tensor_dim0_stride[31:0]
    g1[6] = 0u;
    g1[7] = 0u;
    asm volatile("tensor_load_to_lds %0, %1" :: "s"(g0), "s"(g1) : "memory");
    __builtin_amdgcn_s_wait_tensorcnt(0);  // per-wave: only our slice must land
  }
  // No __syncthreads(): each wave reads only the LDS slice its own TDM wrote.

  // ---- per-thread top-8 over its padded LDS row ----
  const int tid = threadIdx.x;
  const float* row = &smem[tid * ROW_DW];

  uint32_t keyv[NUM_EXPERTS];
#pragma unroll
  for (int i = 0; i < NUM_EXPERTS / 4; ++i) {  // 16x ds_load_b128
    float4 v = reinterpret_cast<const float4*>(row)[i];
    keyv[4 * i + 0] = f32_to_key(v.x);
    keyv[4 * i + 1] = f32_to_key(v.y);
    keyv[4 * i + 2] = f32_to_key(v.z);
    keyv[4 * i + 3] = f32_to_key(v.w);
  }

  float pk[TOPK];
  int ik[TOPK];
  uint32_t prev = 0xFFFFFFFFu;  // no winner masked in round 0
  float vmax = 0.0f, ssum = 0.0f;
#pragma unroll
  for (int r = 0; r < TOPK; ++r) {
    // 4 independent argmax chains (indices 0..15 | 16..31 | 32..47 | 48..63).
    uint32_t bk0 = 0u, bi0 = 0u, bk1 = 0u, bi1 = 0u;
    uint32_t bk2 = 0u, bi2 = 0u, bk3 = 0u, bi3 = 0u;
#pragma unroll
    for (int e = 0; e < 16; ++e) {
      uint32_t k0 = keyv[e];
      uint32_t k1 = keyv[e + 16];
      uint32_t k2 = keyv[e + 32];
      uint32_t k3 = keyv[e + 48];
      k0 = (prev == (uint32_t)(e)) ? 0u : k0;
      k1 = (prev == (uint32_t)(e + 16)) ? 0u : k1;
      k2 = (prev == (uint32_t)(e + 32)) ? 0u : k2;
      k3 = (prev == (uint32_t)(e + 48)) ? 0u : k3;
      keyv[e] = k0;
      keyv[e + 16] = k1;
      keyv[e + 32] = k2;
      keyv[e + 48] = k3;
      if (k0 > bk0) { bk0 = k0; bi0 = (uint32_t)(e); }
      if (k1 > bk1) { bk1 = k1; bi1 = (uint32_t)(e + 16); }
      if (k2 > bk2) { bk2 = k2; bi2 = (uint32_t)(e + 32); }
      if (k3 > bk3) { bk3 = k3; bi3 = (uint32_t)(e + 48); }
    }
    // Merge: left operand always holds lower indices; strict '>' keeps it on
    // ties, preserving lax.top_k's lower-index-first order.
    if (bk1 > bk0) { bk0 = bk1; bi0 = bi1; }
    if (bk3 > bk2) { bk2 = bk3; bi2 = bi3; }
    if (bk2 > bk0) { bk0 = bk2; bi0 = bi2; }

    prev = bi0;
    float v = key_to_f32(bk0);
    if (r == 0) vmax = v;  // round-0 winner is the row max
    float p = __expf(v - vmax);
    ssum += p;
    pk[r] = p;
    ik[r] = (int)bi0;
  }

  const long tok = block_tok0 + tid;
  if (tok < num_tokens) {
    float inv = 1.0f / ssum;
    float4 w0 = make_float4(pk[0] * inv, pk[1] * inv, pk[2] * inv, pk[3] * inv);
    float4 w1 = make_float4(pk[4] * inv, pk[5] * inv, pk[6] * inv, pk[7] * inv);
    reinterpret_cast<float4*>(w_out + tok * TOPK)[0] = w0;
    reinterpret_cast<float4*>(w_out + tok * TOPK)[1] = w1;
    int4 i0 = make_int4(ik[0], ik[1], ik[2], ik[3]);
    int4 i1 = make_int4(ik[4], ik[5], ik[6], ik[7]);
    reinterpret_cast<int4*>(id_out + tok * TOPK)[0] = i0;
    reinterpret_cast<int4*>(id_out + tok * TOPK)[1] = i1;
    int base = (int)(tok * TOPK);
    int4 s0 = make_int4(base + 0, base + 1, base + 2, base + 3);
    int4 s1 = make_int4(base + 4, base + 5, base + 6, base + 7);
    reinterpret_cast<int4*>(src_out + tok * TOPK)[0] = s0;
    reinterpret_cast<int4*>(src_out + tok * TOPK)[1] = s1;
  }
}

extern "C" void kernel_launch(void* const* d_in, const int* in_sizes, int n_in,
                              void* d_out, int out_size, void* d_ws, size_t ws_size,
                              hipStream_t stream) {
  (void)n_in; (void)d_ws; (void)ws_size; (void)out_size;
  const float* logits = (const float*)d_in[0];
  // d_in[1] is top_k (==8), baked into TOPK.
  const long T = (long)in_sizes[0] / NUM_EXPERTS;
  float* w = (float*)d_out;                                  // [T, 8] fp32
  int* ids = (int*)((float*)d_out + (size_t)T * TOPK);       // [T, 8] int32
  int* src = (int*)((float*)d_out + (size_t)2 * T * TOPK);   // [T, 8] int32
  const int grid = (int)((T + TPB - 1) / TPB);
  topk_softmax_kernel<<<grid, TPB, 0, stream>>>(logits, w, ids, src, T);
}